// DoubleGraphGNN_10900626997630
// MI455X (gfx1250) — compile-verified
//
#include <hip/hip_runtime.h>
#include <hip/hip_bf16.h>

typedef float v2f __attribute__((ext_vector_type(2)));
typedef float v8f __attribute__((ext_vector_type(8)));

#define NN_NODES 50000
#define NN_EDGES 600000
#define NF_IN    64
#define NN_GRAPH 2000
#define BN_EPS   1e-5f
#define BN_ROWS_PER_BLOCK 128

__global__ void fill_f32(float* __restrict__ p, float v, int n) {
  int i = blockIdx.x * blockDim.x + threadIdx.x;
  if (i < n) p[i] = v;
}

__global__ void deg_count(const int* __restrict__ dst, float* __restrict__ deg, int E) {
  int e = blockIdx.x * blockDim.x + threadIdx.x;
  if (e < E) atomicAdd(&deg[dst[e]], 1.0f);
}

// C[MxN] = A[MxK] * W[KxN] (+ bias). One wave per 16x16 tile, f32 WMMA, K-step 4.
// grid = (M/16, N/16), block = 32 (one wave; EXEC all ones, no divergence).
__global__ void wmma_gemm_f32(const float* __restrict__ A, const float* __restrict__ W,
                              float* __restrict__ C, const float* __restrict__ bias,
                              int M, int K, int N) {
  const int m0   = blockIdx.x * 16;
  const int n0   = blockIdx.y * 16;
  const int lane = threadIdx.x;
  const int half = lane >> 4;   // 0: lanes 0-15, 1: lanes 16-31
  const int l15  = lane & 15;
  v8f acc = {0.f, 0.f, 0.f, 0.f, 0.f, 0.f, 0.f, 0.f};
  const float* arow = A + (size_t)(m0 + l15) * K;
  for (int k0 = 0; k0 < K; k0 += 4) {
    const int ak = k0 + half * 2;           // K index pair base for this lane half
    v2f a, b;
    a.x = arow[ak];                          // A: (M=l15, K=ak)
    a.y = arow[ak + 1];                      // A: (M=l15, K=ak+1)
    b.x = W[(size_t)ak * N + n0 + l15];      // B: (K=ak,   N=l15)
    b.y = W[(size_t)(ak + 1) * N + n0 + l15];// B: (K=ak+1, N=l15)
    acc = __builtin_amdgcn_wmma_f32_16x16x4_f32(
        /*neg_a=*/false, a, /*neg_b=*/false, b,
        /*c_mod=*/(short)0, acc, /*reuse_a=*/false, /*reuse_b=*/false);
  }
  const float bv = bias ? bias[n0 + l15] : 0.0f;
#pragma unroll
  for (int j = 0; j < 8; ++j) {
    int row = j + half * 8;                  // C/D layout: VGPR j -> M = j (+8 for hi half)
    C[(size_t)(m0 + row) * N + n0 + l15] = acc[j] + bv;
  }
}

// agg = h * (1/deg) + bias   (self-loop term + conv bias; scatter accumulates on top)
__global__ void agg_init(const float* __restrict__ h, const float* __restrict__ deg,
                         const float* __restrict__ bias, float* __restrict__ agg,
                         int Nrows, int Cf) {
  int idx = blockIdx.x * blockDim.x + threadIdx.x;
  if (idx >= Nrows * Cf) return;
  int r = idx / Cf, c = idx - r * Cf;
  agg[idx] = h[idx] / deg[r] + bias[c];
}

// agg[dst] += h[src] * rsqrt(deg[src]) * rsqrt(deg[dst]); one thread per (edge, feature)
__global__ void edge_scatter(const float* __restrict__ h, const int* __restrict__ src,
                             const int* __restrict__ dst, const float* __restrict__ deg,
                             float* __restrict__ agg, int E, int Cf) {
  int idx = blockIdx.x * blockDim.x + threadIdx.x;
  if (idx >= E * Cf) return;
  int e = idx / Cf, c = idx - e * Cf;
  int s = src[e], d = dst[e];
  float norm = rsqrtf(deg[s]) * rsqrtf(deg[d]);
  atomicAdd(&agg[d * Cf + c], h[s * Cf + c] * norm);
}

// per-feature sum and sum-of-squares: stats[0..Cf-1]=sum, stats[Cf..2Cf-1]=sumsq
__global__ void bn_stats(const float* __restrict__ x, float* __restrict__ stats,
                         int Nrows, int Cf) {
  int c = threadIdx.x;                       // blockDim.x == Cf
  int r0 = blockIdx.x * BN_ROWS_PER_BLOCK;
  int rend = min(r0 + BN_ROWS_PER_BLOCK, Nrows);
  float s = 0.f, s2 = 0.f;
  for (int r = r0; r < rend; ++r) {
    float v = x[(size_t)r * Cf + c];
    s += v; s2 += v * v;
  }
  atomicAdd(&stats[c], s);
  atomicAdd(&stats[Cf + c], s2);
}

__global__ void bn_relu(float* __restrict__ x, const float* __restrict__ stats,
                        const float* __restrict__ gamma, const float* __restrict__ beta,
                        int Nrows, int Cf) {
  int idx = blockIdx.x * blockDim.x + threadIdx.x;
  if (idx >= Nrows * Cf) return;
  int c = idx % Cf;
  float inv_n = 1.0f / (float)Nrows;
  float m   = stats[c] * inv_n;
  float var = stats[Cf + c] * inv_n - m * m;  // biased var (matches jnp.var)
  float y = (x[idx] - m) * rsqrtf(var + BN_EPS) * gamma[c] + beta[c];
  x[idx] = fmaxf(y, 0.0f);
}

__global__ void count_nodes(const int* __restrict__ batch, float* __restrict__ gcount, int Nrows) {
  int i = blockIdx.x * blockDim.x + threadIdx.x;
  if (i < Nrows) atomicAdd(&gcount[batch[i]], 1.0f);
}

__global__ void pool_sum(const float* __restrict__ x, const int* __restrict__ batch,
                         float* __restrict__ pooled, int colOff, int ldp, int Nrows, int Cf) {
  int idx = blockIdx.x * blockDim.x + threadIdx.x;
  if (idx >= Nrows * Cf) return;
  int r = idx / Cf, c = idx - r * Cf;
  atomicAdd(&pooled[batch[r] * ldp + colOff + c], x[idx]);
}

__global__ void pool_div(float* __restrict__ pooled, const float* __restrict__ gcount,
                         int G, int ldp) {
  int idx = blockIdx.x * blockDim.x + threadIdx.x;
  if (idx >= G * ldp) return;
  pooled[idx] /= fmaxf(gcount[idx / ldp], 1.0f);
}

__global__ void head_out(const float* __restrict__ h, const float* __restrict__ Wf2,
                         const float* __restrict__ bf2, float* __restrict__ out, int G) {
  int g = blockIdx.x * blockDim.x + threadIdx.x;
  if (g >= G) return;
  float acc = bf2[0];
#pragma unroll
  for (int c = 0; c < 64; ++c) acc += h[g * 64 + c] * Wf2[c];
  out[g] = acc;
}

extern "C" void kernel_launch(void* const* d_in, const int* in_sizes, int n_in,
                              void* d_out, int out_size, void* d_ws, size_t ws_size,
                              hipStream_t stream) {
  (void)in_sizes; (void)n_in; (void)out_size; (void)ws_size;
  const int E = NN_EDGES, Nn = NN_NODES, G = NN_GRAPH;
  const int* batch = (const int*)d_in[4];

  // workspace layout (floats): ~53 MB total
  float* ws = (float*)d_ws;
  size_t off = 0;
  float* buf0   = ws + off; off += (size_t)Nn * 128;  // h1; later h2 (lo half) + agg2 (hi half)
  float* buf1   = ws + off; off += (size_t)Nn * 128;  // agg1 (BN in-place)
  float* deg    = ws + off; off += Nn;
  float* stats  = ws + off; off += 256;
  float* pooled = ws + off; off += (size_t)G * 128;   // head input [G,128]
  float* gcount = ws + off; off += G;
  float* headh  = ws + off; off += (size_t)G * 64;

  auto cdiv = [](int a, int b) { return (a + b - 1) / b; };

  fill_f32<<<cdiv(G * 128, 256), 256, 0, stream>>>(pooled, 0.f, G * 128);
  fill_f32<<<cdiv(G, 256), 256, 0, stream>>>(gcount, 0.f, G);
  count_nodes<<<cdiv(Nn, 256), 256, 0, stream>>>(batch, gcount, Nn);

  for (int b = 0; b < 2; ++b) {  // 0 = chromophore, 1 = solvent
    const float* x  = (const float*)d_in[b];
    const int* src  = (const int*)d_in[2 + b];
    const int* dst  = src + E;
    const int pb = 5 + b * 8;
    const float* W1  = (const float*)d_in[pb + 0];
    const float* b1  = (const float*)d_in[pb + 1];
    const float* g1  = (const float*)d_in[pb + 2];
    const float* be1 = (const float*)d_in[pb + 3];
    const float* W2  = (const float*)d_in[pb + 4];
    const float* b2  = (const float*)d_in[pb + 5];
    const float* g2  = (const float*)d_in[pb + 6];
    const float* be2 = (const float*)d_in[pb + 7];

    // degrees (with +1 self-loop)
    fill_f32<<<cdiv(Nn, 256), 256, 0, stream>>>(deg, 1.0f, Nn);
    deg_count<<<cdiv(E, 256), 256, 0, stream>>>(dst, deg, E);

    // ---- GCN layer 1: 64 -> 128 ----
    wmma_gemm_f32<<<dim3(Nn / 16, 128 / 16), 32, 0, stream>>>(x, W1, buf0, nullptr, Nn, NF_IN, 128);
    agg_init<<<cdiv(Nn * 128, 256), 256, 0, stream>>>(buf0, deg, b1, buf1, Nn, 128);
    edge_scatter<<<cdiv(E * 128, 256), 256, 0, stream>>>(buf0, src, dst, deg, buf1, E, 128);
    fill_f32<<<1, 256, 0, stream>>>(stats, 0.f, 256);
    bn_stats<<<cdiv(Nn, BN_ROWS_PER_BLOCK), 128, 0, stream>>>(buf1, stats, Nn, 128);
    bn_relu<<<cdiv(Nn * 128, 256), 256, 0, stream>>>(buf1, stats, g1, be1, Nn, 128);

    // ---- GCN layer 2: 128 -> 64 ----
    float* h2   = buf0;
    float* agg2 = buf0 + (size_t)Nn * 64;
    wmma_gemm_f32<<<dim3(Nn / 16, 64 / 16), 32, 0, stream>>>(buf1, W2, h2, nullptr, Nn, 128, 64);
    agg_init<<<cdiv(Nn * 64, 256), 256, 0, stream>>>(h2, deg, b2, agg2, Nn, 64);
    edge_scatter<<<cdiv(E * 64, 256), 256, 0, stream>>>(h2, src, dst, deg, agg2, E, 64);
    fill_f32<<<1, 256, 0, stream>>>(stats, 0.f, 256);
    bn_stats<<<cdiv(Nn, BN_ROWS_PER_BLOCK), 64, 0, stream>>>(agg2, stats, Nn, 64);
    bn_relu<<<cdiv(Nn * 64, 256), 256, 0, stream>>>(agg2, stats, g2, be2, Nn, 64);

    // ---- mean pool (sum phase; writes branch's 64-column slice of [G,128]) ----
    pool_sum<<<cdiv(Nn * 64, 256), 256, 0, stream>>>(agg2, batch, pooled, b * 64, 128, Nn, 64);
  }

  pool_div<<<cdiv(G * 128, 256), 256, 0, stream>>>(pooled, gcount, G, 128);

  // ---- MLP head ----
  const float* Wf1  = (const float*)d_in[21];
  const float* bf1  = (const float*)d_in[22];
  const float* gf1  = (const float*)d_in[23];
  const float* bef1 = (const float*)d_in[24];
  const float* Wf2  = (const float*)d_in[25];
  const float* bf2  = (const float*)d_in[26];

  wmma_gemm_f32<<<dim3(G / 16, 64 / 16), 32, 0, stream>>>(pooled, Wf1, headh, bf1, G, 128, 64);
  fill_f32<<<1, 256, 0, stream>>>(stats, 0.f, 256);
  bn_stats<<<cdiv(G, BN_ROWS_PER_BLOCK), 64, 0, stream>>>(headh, stats, G, 64);
  bn_relu<<<cdiv(G * 64, 256), 256, 0, stream>>>(headh, stats, gf1, bef1, G, 64);
  head_out<<<cdiv(G, 256), 256, 0, stream>>>(headh, Wf2, bf2, (float*)d_out, G);
}